// GCN_model_27006754357501
// MI455X (gfx1250) — compile-verified
//
#include <hip/hip_runtime.h>
#include <hip/hip_bf16.h>
#include <cstddef>

typedef __attribute__((ext_vector_type(2))) float v2f;
typedef __attribute__((ext_vector_type(8))) float v8f;

#define NN 2228224      // 131072 * 17 nodes
#define EE 4194304      // edges
#define BB 131072       // graphs

// native fp32 atomic add (global_atomic_add_f32, no CAS loop)
__device__ __forceinline__ void atomAddF(float* p, float v) {
    unsafeAtomicAdd(p, v);
}

// ---------------- degree / normalization ----------------
__global__ void k_ones(float* deg, int n) {
    int i = blockIdx.x * 256 + threadIdx.x;
    if (i < n) deg[i] = 1.0f;                 // self-loop contribution
}

__global__ void k_degadd(const long long* __restrict__ ei, float* deg, int E) {
    int e = blockIdx.x * 256 + threadIdx.x;
    if (e < E) atomAddF(&deg[ei[(size_t)E + e]], 1.0f);   // dst row
}

__global__ void k_rsqrt(float* deg, int n) {
    int i = blockIdx.x * 256 + threadIdx.x;
    if (i < n) deg[i] = rsqrtf(deg[i]);       // deg >= 1 always
}

__global__ void k_zero(float* p, int n) {
    int i = blockIdx.x * 256 + threadIdx.x;
    if (i < n) p[i] = 0.0f;
}

// ---------------- layer 1: self-loop init + edge scatter (x@W1 fused) ----------------
__global__ void k_self1(const float* __restrict__ x, const float* __restrict__ W1,
                        const float* __restrict__ b1, const float* __restrict__ dinv,
                        float* __restrict__ o1, int n) {
    int i = blockIdx.x * 256 + threadIdx.x;
    if (i >= n) return;
    float di = dinv[i], nrm = di * di;
    const float* xs = x + (size_t)i * 3;
    float x0 = xs[0] * nrm, x1 = xs[1] * nrm, x2 = xs[2] * nrm;
    float* o = o1 + (size_t)i * 8;
#pragma unroll
    for (int c = 0; c < 8; ++c)
        o[c] = x0 * W1[c] + x1 * W1[8 + c] + x2 * W1[16 + c] + b1[c];
}

__global__ void k_edge1(const long long* __restrict__ ei, const float* __restrict__ x,
                        const float* __restrict__ W1, const float* __restrict__ dinv,
                        float* __restrict__ o1, int E) {
    __shared__ float sW[24];
    if (threadIdx.x < 24) sW[threadIdx.x] = W1[threadIdx.x];
    __syncthreads();
    int e = blockIdx.x * 256 + threadIdx.x;
    if (e >= E) return;
    long long s = ei[e], d = ei[(size_t)E + e];
    float nrm = dinv[s] * dinv[d];
    const float* xs = x + (size_t)s * 3;
    float x0 = xs[0] * nrm, x1 = xs[1] * nrm, x2 = xs[2] * nrm;
    float* od = o1 + (size_t)d * 8;
#pragma unroll
    for (int c = 0; c < 8; ++c)
        atomAddF(&od[c], x0 * sW[c] + x1 * sW[8 + c] + x2 * sW[16 + c]);
}

// ---------------- layer 2: self-loop init + edge scatter (a1@W2 fused) ----------------
__global__ void k_self2(const float* __restrict__ a1, const float* __restrict__ W2,
                        const float* __restrict__ b2, const float* __restrict__ dinv,
                        float* __restrict__ o2, int n) {
    int i = blockIdx.x * 256 + threadIdx.x;
    if (i >= n) return;
    float di = dinv[i], nrm = di * di;
    const float* as = a1 + (size_t)i * 8;
    float av[8];
#pragma unroll
    for (int k = 0; k < 8; ++k) av[k] = as[k] * nrm;
    float* o = o2 + (size_t)i * 16;
#pragma unroll
    for (int c = 0; c < 16; ++c) {
        float m = b2[c];
#pragma unroll
        for (int k = 0; k < 8; ++k) m += av[k] * W2[k * 16 + c];
        o[c] = m;
    }
}

__global__ void k_edge2(const long long* __restrict__ ei, const float* __restrict__ a1,
                        const float* __restrict__ W2, const float* __restrict__ dinv,
                        float* __restrict__ o2, int E) {
    __shared__ float sW[128];
    if (threadIdx.x < 128) sW[threadIdx.x] = W2[threadIdx.x];
    __syncthreads();
    int e = blockIdx.x * 256 + threadIdx.x;
    if (e >= E) return;
    long long s = ei[e], d = ei[(size_t)E + e];
    float nrm = dinv[s] * dinv[d];
    const float* as = a1 + (size_t)s * 8;
    float av[8];
#pragma unroll
    for (int k = 0; k < 8; ++k) av[k] = as[k] * nrm;
    float* od = o2 + (size_t)d * 16;
#pragma unroll
    for (int c = 0; c < 16; ++c) {
        float m = 0.0f;
#pragma unroll
        for (int k = 0; k < 8; ++k) m += av[k] * sW[k * 16 + c];
        atomAddF(&od[c], m);
    }
}

// ---------------- batchnorm: column sums, finalize, apply+relu ----------------
template <int C>
__global__ void k_bnstats(const float* __restrict__ d, long long total,
                          float* __restrict__ sums, float* __restrict__ sqs) {
    __shared__ float ssum[256], ssq[256];
    int tid = threadIdx.x;
    float s = 0.0f, q = 0.0f;
    for (long long i = (long long)blockIdx.x * 256 + tid; i < total;
         i += (long long)gridDim.x * 256) {
        float v = d[i];
        s += v; q += v * v;
    }
    ssum[tid] = s; ssq[tid] = q;
    __syncthreads();
    for (int st = 128; st >= C; st >>= 1) {
        if (tid < st) { ssum[tid] += ssum[tid + st]; ssq[tid] += ssq[tid + st]; }
        __syncthreads();
    }
    if (tid < C) { atomAddF(&sums[tid], ssum[tid]); atomAddF(&sqs[tid], ssq[tid]); }
}

__global__ void k_bnfinal(const float* __restrict__ sums, const float* __restrict__ sqs,
                          const float* __restrict__ g, const float* __restrict__ be,
                          float* __restrict__ scale, float* __restrict__ shift,
                          int C, float invN) {
    int c = threadIdx.x;
    if (c >= C) return;
    float mean = sums[c] * invN;
    float var  = sqs[c] * invN - mean * mean;
    float sc   = g[c] * rsqrtf(var + 1e-5f);
    scale[c] = sc;
    shift[c] = be[c] - mean * sc;
}

__global__ void k_bnapply(float* __restrict__ d, long long total,
                          const float* __restrict__ scale, const float* __restrict__ shift,
                          int cmask) {
    long long i = (long long)blockIdx.x * 256 + threadIdx.x;
    if (i >= total) return;
    int c = (int)i & cmask;
    float v = d[i] * scale[c] + shift[c];
    d[i] = v > 0.0f ? v : 0.0f;
}

// ---------------- readout: [B,272]@[272,32] via V_WMMA_F32_16X16X4_F32, fused relu + @Wf2 ----------------
// LDS holds Wf1 pair-interleaved: sW[pair][col][parity], pair = row/2, so each
// B fragment (rows 2p,2p+1 at one column) is a single aligned ds_load_b64.
__global__ __launch_bounds__(256) void k_fc(const float* __restrict__ A,
                                            const float* __restrict__ Wf1,
                                            const float* __restrict__ bf1,
                                            const float* __restrict__ Wf2,
                                            const float* __restrict__ bf2,
                                            float* __restrict__ out) {
    __shared__ float sW[136 * 64];               // 34 KB: 272 rows x 32 cols
    for (int i = threadIdx.x; i < 272 * 32; i += 256) {
        int r = i >> 5, c = i & 31;
        sW[(r >> 1) * 64 + c * 2 + (r & 1)] = Wf1[i];
    }
    __syncthreads();

    int wave = threadIdx.x >> 5;
    int lane = threadIdx.x & 31;
    int row0 = (blockIdx.x * 8 + wave) * 16;     // 16 batch rows per wave
    int m  = lane & 15;                           // A: row / B,D: column
    int kh = (lane >> 4) << 1;                    // lane-half K offset: 0 or 2
    const float* arow = A + (size_t)(row0 + m) * 272 + kh;

    v8f acc0 = {0.f, 0.f, 0.f, 0.f, 0.f, 0.f, 0.f, 0.f};
    v8f acc1 = acc0;
    for (int k = 0; k < 272; k += 4) {
        float2 av = *reinterpret_cast<const float2*>(arow + k);
        v2f a;  a.x = av.x;  a.y = av.y;          // A[m, k+kh], A[m, k+kh+1]
        int pbase = ((k + kh) >> 1) * 64;         // (k+kh) even
        float2 bv0 = *reinterpret_cast<const float2*>(&sW[pbase + m * 2]);
        float2 bv1 = *reinterpret_cast<const float2*>(&sW[pbase + (m + 16) * 2]);
        v2f b0; b0.x = bv0.x; b0.y = bv0.y;       // B[k+kh, m],    B[k+kh+1, m]
        v2f b1; b1.x = bv1.x; b1.y = bv1.y;       // B[k+kh, m+16], B[k+kh+1, m+16]
        acc0 = __builtin_amdgcn_wmma_f32_16x16x4_f32(false, a, false, b0, (short)0, acc0, false, false);
        acc1 = __builtin_amdgcn_wmma_f32_16x16x4_f32(false, a, false, b1, (short)0, acc1, false, false);
    }

    // epilogue: bias + relu, dot with Wf2[32], reduce 16 lanes per half
    float ba = bf1[m], bb = bf1[m + 16];
    float wa = Wf2[m], wb = Wf2[m + 16];
    float c2 = bf2[0];
    int rbase = row0 + ((lane >> 4) << 3);        // D: VGPR r holds M = r + 8*(lane>>4)
#pragma unroll
    for (int r = 0; r < 8; ++r) {
        float za = acc0[r] + ba; za = za > 0.0f ? za : 0.0f;
        float zb = acc1[r] + bb; zb = zb > 0.0f ? zb : 0.0f;
        float p = za * wa + zb * wb;
        p += __shfl_xor(p, 1);
        p += __shfl_xor(p, 2);
        p += __shfl_xor(p, 4);
        p += __shfl_xor(p, 8);
        if (m == 0) out[rbase + r] = p + c2;
    }
}

extern "C" void kernel_launch(void* const* d_in, const int* in_sizes, int n_in,
                              void* d_out, int out_size, void* d_ws, size_t ws_size,
                              hipStream_t stream) {
    (void)in_sizes; (void)n_in; (void)out_size; (void)ws_size;
    const float*     x   = (const float*)d_in[0];
    const long long* ei  = (const long long*)d_in[1];   // int64 [2, E]
    const float*     W1  = (const float*)d_in[2];
    const float*     b1  = (const float*)d_in[3];
    const float*     g1  = (const float*)d_in[4];
    const float*     be1 = (const float*)d_in[5];
    const float*     W2  = (const float*)d_in[6];
    const float*     b2  = (const float*)d_in[7];
    const float*     g2  = (const float*)d_in[8];
    const float*     be2 = (const float*)d_in[9];
    const float*     Wf1 = (const float*)d_in[10];
    const float*     bf1 = (const float*)d_in[11];
    const float*     Wf2 = (const float*)d_in[12];
    const float*     bf2 = (const float*)d_in[13];
    float* out = (float*)d_out;

    // workspace layout
    float* w     = (float*)d_ws;
    float* dinv  = w;                              // N
    float* o1    = dinv + (size_t)NN;              // N*8  (becomes a1 in place)
    float* o2    = o1 + (size_t)NN * 8;            // N*16 (becomes a2 in place)
    float* stats = o2 + (size_t)NN * 16;           // 96 floats
    float* sum1 = stats, *sq1 = stats + 8,  *sc1 = stats + 16, *sh1 = stats + 24;
    float* sum2 = stats + 32, *sq2 = stats + 48, *sc2 = stats + 64, *sh2 = stats + 80;

    const int nb  = NN / 256;                      // 8704
    const int ebn = EE / 256;                      // 16384
    const float invN = 1.0f / (float)NN;

    // stats accumulators must start at zero every call
    k_zero<<<1, 128, 0, stream>>>(stats, 96);

    // degree -> dinv
    k_ones  <<<nb, 256, 0, stream>>>(dinv, NN);
    k_degadd<<<ebn, 256, 0, stream>>>(ei, dinv, EE);
    k_rsqrt <<<nb, 256, 0, stream>>>(dinv, NN);

    // layer 1: GCNConv(3->8)
    k_self1<<<nb, 256, 0, stream>>>(x, W1, b1, dinv, o1, NN);
    k_edge1<<<ebn, 256, 0, stream>>>(ei, x, W1, dinv, o1, EE);
    k_bnstats<8><<<2048, 256, 0, stream>>>(o1, (long long)NN * 8, sum1, sq1);
    k_bnfinal<<<1, 32, 0, stream>>>(sum1, sq1, g1, be1, sc1, sh1, 8, invN);
    k_bnapply<<<NN * 8 / 256, 256, 0, stream>>>(o1, (long long)NN * 8, sc1, sh1, 7);

    // layer 2: GCNConv(8->16)
    k_self2<<<nb, 256, 0, stream>>>(o1, W2, b2, dinv, o2, NN);
    k_edge2<<<ebn, 256, 0, stream>>>(ei, o1, W2, dinv, o2, EE);
    k_bnstats<16><<<2048, 256, 0, stream>>>(o2, (long long)NN * 16, sum2, sq2);
    k_bnfinal<<<1, 32, 0, stream>>>(sum2, sq2, g2, be2, sc2, sh2, 16, invN);
    k_bnapply<<<NN * 16 / 256, 256, 0, stream>>>(o2, (long long)NN * 16, sc2, sh2, 15);

    // readout: [B,272]@Wf1 (WMMA fp32) + relu, then @Wf2 + bf2 -> [B,1]
    k_fc<<<BB / 128, 256, 0, stream>>>(o2, Wf1, bf1, Wf2, bf2, out);
}